// GCN_61701500174370
// MI455X (gfx1250) — compile-verified
//
#include <hip/hip_runtime.h>

typedef __attribute__((ext_vector_type(2))) float v2f;
typedef __attribute__((ext_vector_type(8))) float v8f;

__device__ __forceinline__ float lrelu(float v) { return v > 0.0f ? v : 0.01f * v; }

// ---------------- gcn_norm ----------------
__global__ void k_deg_init(float* __restrict__ deg, int N) {
  int i = blockIdx.x * blockDim.x + threadIdx.x;
  if (i < N) deg[i] = 1.0f;  // self-loop weight contributes 1 to its own target degree
}

__global__ void k_deg_accum(const long long* __restrict__ col,
                            const float* __restrict__ w,
                            float* __restrict__ deg, int E) {
  int e = blockIdx.x * blockDim.x + threadIdx.x;
  if (e < E) atomicAdd(&deg[col[e]], w[e]);
}

__global__ void k_dinv(float* __restrict__ deg, int N) {
  int i = blockIdx.x * blockDim.x + threadIdx.x;
  if (i < N) {
    float d = deg[i];
    deg[i] = d > 0.0f ? rsqrtf(d) : 0.0f;
  }
}

// ---------------- dense transforms via V_WMMA_F32_16X16X4_F32 ----------------
// T[N,32] = X[N,128] @ W[128,32]; one wave -> 16x32 tile, K walked 4 at a time.
__global__ __launch_bounds__(128) void k_gemm1(const float* __restrict__ X,
                                               const float* __restrict__ W,
                                               float* __restrict__ T, int N) {
  const int lane = threadIdx.x & 31;
  const int wv = threadIdx.x >> 5;
  const long tile = (long)blockIdx.x * 4 + wv;
  if (tile * 16 >= N) return;  // wave-uniform guard: EXEC stays all-ones for WMMA
  const int m = lane & 15;
  const int kh = (lane >> 4) << 1;  // 0 for lanes 0-15, 2 for lanes 16-31
  const long r0 = tile * 16;
  const float* xrow = X + (r0 + m) * 128 + kh;
  v8f acc0 = {};
  v8f acc1 = {};
#pragma unroll
  for (int kk = 0; kk < 128; kk += 4) {
    v2f a, b0, b1;
    a.x = xrow[kk];
    a.y = xrow[kk + 1];
    const float* w0 = W + (kk + kh) * 32;  // rows K=kk+kh and kk+kh+1 of W
    b0.x = w0[m];        // B[K=kh][n=m]        (cols 0..15)
    b0.y = w0[32 + m];   // B[K=kh+1][n=m]
    b1.x = w0[16 + m];   // cols 16..31
    b1.y = w0[48 + m];
    acc0 = __builtin_amdgcn_wmma_f32_16x16x4_f32(false, a, false, b0, (short)0, acc0, false, false);
    acc1 = __builtin_amdgcn_wmma_f32_16x16x4_f32(false, a, false, b1, (short)0, acc1, false, false);
  }
  // C/D layout: VGPR v, lanes 0-15 -> M=v, lanes 16-31 -> M=8+v; N=lane&15
  float* out = T + (r0 + ((lane >> 4) << 3)) * 32 + m;
#pragma unroll
  for (int v = 0; v < 8; ++v) {
    out[v * 32] = acc0[v];
    out[v * 32 + 16] = acc1[v];
  }
}

// T[N,16] = H[N,32] @ W[32,16]; one wave -> 16x16 tile, 8 WMMA steps.
__global__ __launch_bounds__(128) void k_gemm2(const float* __restrict__ H,
                                               const float* __restrict__ W,
                                               float* __restrict__ T, int N) {
  const int lane = threadIdx.x & 31;
  const int wv = threadIdx.x >> 5;
  const long tile = (long)blockIdx.x * 4 + wv;
  if (tile * 16 >= N) return;
  const int m = lane & 15;
  const int kh = (lane >> 4) << 1;
  const long r0 = tile * 16;
  const float* hrow = H + (r0 + m) * 32 + kh;
  v8f acc = {};
#pragma unroll
  for (int kk = 0; kk < 32; kk += 4) {
    v2f a, b;
    a.x = hrow[kk];
    a.y = hrow[kk + 1];
    const float* w0 = W + (kk + kh) * 16;
    b.x = w0[m];
    b.y = w0[16 + m];
    acc = __builtin_amdgcn_wmma_f32_16x16x4_f32(false, a, false, b, (short)0, acc, false, false);
  }
  float* out = T + (r0 + ((lane >> 4) << 3)) * 16 + m;
#pragma unroll
  for (int v = 0; v < 8; ++v) out[v * 16] = acc[v];
}

// ---------------- aggregation ----------------
// Init agg with the self-loop term dinv[i]^2 * t[i,:] (no atomics needed).
__global__ void k_self_init(const float* __restrict__ t, const float* __restrict__ dinv,
                            float* __restrict__ agg, long total, int logF) {
  long idx = (long)blockIdx.x * blockDim.x + threadIdx.x;
  if (idx < total) {
    float d = dinv[idx >> logF];
    agg[idx] = t[idx] * d * d;
  }
}

// One wave per edge; lane = feature (32 features). 128B contiguous atomic burst per edge.
__global__ void k_scatter32(const long long* __restrict__ row, const long long* __restrict__ col,
                            const float* __restrict__ w, const float* __restrict__ dinv,
                            const float* __restrict__ t, float* __restrict__ agg, int E) {
  int gid = blockIdx.x * blockDim.x + threadIdx.x;
  int e = gid >> 5;
  int f = gid & 31;
  if (e >= E) return;
  long long r = row[e];
  long long c = col[e];
  float nw = dinv[r] * w[e] * dinv[c];
  atomicAdd(&agg[c * 32 + f], t[r * 32 + f] * nw);
}

// Half-wave per edge; lane%16 = feature (16 features).
__global__ void k_scatter16(const long long* __restrict__ row, const long long* __restrict__ col,
                            const float* __restrict__ w, const float* __restrict__ dinv,
                            const float* __restrict__ t, float* __restrict__ agg, int E) {
  int gid = blockIdx.x * blockDim.x + threadIdx.x;
  int e = gid >> 4;
  int f = gid & 15;
  if (e >= E) return;
  long long r = row[e];
  long long c = col[e];
  float nw = dinv[r] * w[e] * dinv[c];
  atomicAdd(&agg[c * 16 + f], t[r * 16 + f] * nw);
}

__global__ void k_bias_lrelu(float* __restrict__ a, const float* __restrict__ b,
                             long total, int fmask) {
  long idx = (long)blockIdx.x * blockDim.x + threadIdx.x;
  if (idx < total) a[idx] = lrelu(a[idx] + b[idx & fmask]);
}

// Fused: h2 = lrelu(agg2 + b2); out = h2 @ Wfc + bfc
__global__ void k_fc(const float* __restrict__ agg, const float* __restrict__ b2,
                     const float* __restrict__ Wfc, const float* __restrict__ bfc,
                     float* __restrict__ out, int N) {
  int i = blockIdx.x * blockDim.x + threadIdx.x;
  if (i >= N) return;
  const float* rowp = agg + (long)i * 16;
  float s = 0.0f;
#pragma unroll
  for (int f = 0; f < 16; ++f) s += lrelu(rowp[f] + b2[f]) * Wfc[f];
  out[i] = s + bfc[0];
}

extern "C" void kernel_launch(void* const* d_in, const int* in_sizes, int n_in,
                              void* d_out, int out_size, void* d_ws, size_t ws_size,
                              hipStream_t stream) {
  const float* x = (const float*)d_in[0];
  const long long* ei = (const long long*)d_in[1];
  const float* ew = (const float*)d_in[2];
  const float* W1 = (const float*)d_in[3];
  const float* b1 = (const float*)d_in[4];
  const float* W2 = (const float*)d_in[5];
  const float* b2 = (const float*)d_in[6];
  const float* Wfc = (const float*)d_in[7];
  const float* bfc = (const float*)d_in[8];
  float* out = (float*)d_out;

  const int N = in_sizes[0] / 128;
  const int E = in_sizes[2];
  const long long* row = ei;       // edge_index[0]
  const long long* col = ei + E;   // edge_index[1]

  float* ws = (float*)d_ws;
  float* dinv = ws;                       // N      (deg -> dinv in place)
  float* t1 = dinv + N;                   // N*32   (x@W1)
  float* agg1 = t1 + (size_t)N * 32;      // N*32   (aggregated -> h1 in place)
  float* t2 = agg1 + (size_t)N * 32;      // N*16   (h1@W2)
  float* agg2 = t2 + (size_t)N * 16;      // N*16

  const int TB = 256;
  // normalization
  k_deg_init<<<(N + TB - 1) / TB, TB, 0, stream>>>(dinv, N);
  k_deg_accum<<<(E + TB - 1) / TB, TB, 0, stream>>>(col, ew, dinv, E);
  k_dinv<<<(N + TB - 1) / TB, TB, 0, stream>>>(dinv, N);

  const int tiles = (N + 15) / 16;
  const unsigned gemmBlocks = (unsigned)((tiles + 3) / 4);

  // layer 1
  k_gemm1<<<gemmBlocks, 128, 0, stream>>>(x, W1, t1, N);
  long tot1 = (long)N * 32;
  k_self_init<<<(unsigned)((tot1 + TB - 1) / TB), TB, 0, stream>>>(t1, dinv, agg1, tot1, 5);
  k_scatter32<<<(unsigned)((E + 7) / 8), TB, 0, stream>>>(row, col, ew, dinv, t1, agg1, E);
  k_bias_lrelu<<<(unsigned)((tot1 + TB - 1) / TB), TB, 0, stream>>>(agg1, b1, tot1, 31);

  // layer 2
  k_gemm2<<<gemmBlocks, 128, 0, stream>>>(agg1, W2, t2, N);
  long tot2 = (long)N * 16;
  k_self_init<<<(unsigned)((tot2 + TB - 1) / TB), TB, 0, stream>>>(t2, dinv, agg2, tot2, 4);
  k_scatter16<<<(unsigned)((E + 15) / 16), TB, 0, stream>>>(row, col, ew, dinv, t2, agg2, E);

  // fused bias + leaky-relu + FC head
  k_fc<<<(N + TB - 1) / TB, TB, 0, stream>>>(agg2, b2, Wfc, bfc, out, N);
}